// HybridSelfAttentionEstimator_65481071395386
// MI455X (gfx1250) — compile-verified
//
#include <hip/hip_runtime.h>
#include <math.h>

#define BATCH   1024
#define SEQ     256
#define EMBED   4
#define H1D     8
#define H2D     4
#define NWAVE   4          // 128 threads per block
#define SSTRIDE 260        // score row stride (dwords); 260 % 64 == 4 -> conflict-free WMMA stores

typedef __attribute__((ext_vector_type(2))) float v2f;
typedef __attribute__((ext_vector_type(8))) float v8f;

__global__ __launch_bounds__(128)
void hybrid_attn_kernel(const float* __restrict__ x,
                        const float* __restrict__ Wq, const float* __restrict__ Wk,
                        const float* __restrict__ Wv,
                        const float* __restrict__ W1, const float* __restrict__ b1,
                        const float* __restrict__ W2, const float* __restrict__ b2,
                        const float* __restrict__ W3, const float* __restrict__ b3,
                        float* __restrict__ out)
{
  extern __shared__ float smem[];
  float*  q_lds = smem;                           // 256*4 (holds 0.5*q: scale pre-folded)
  float*  k_lds = q_lds + SEQ * EMBED;            // 256*4
  float4* v4    = (float4*)(k_lds + SEQ * EMBED); // 256 float4
  float*  sbuf  = (float*)(v4 + SEQ);             // NWAVE * 16 * SSTRIDE

  const int b    = blockIdx.x;
  const int tid  = threadIdx.x;
  const int lane = tid & 31;
  const int wave = tid >> 5;

  // ---- Phase 1: q/k/v projections (E=4) into LDS; 1/sqrt(E) folded into q ----
  const float* xb = x + (size_t)b * SEQ * EMBED;
  for (int t = tid; t < SEQ; t += 128) {
    float4 xe = ((const float4*)xb)[t];
    float xv[4] = {xe.x, xe.y, xe.z, xe.w};
    float4 vvec;
    float* vp = (float*)&vvec;
#pragma unroll
    for (int eo = 0; eo < 4; ++eo) {
      float q = 0.f, k = 0.f, v = 0.f;
#pragma unroll
      for (int e = 0; e < 4; ++e) {
        q += xv[e] * Wq[e * 4 + eo];
        k += xv[e] * Wk[e * 4 + eo];
        v += xv[e] * Wv[e * 4 + eo];
      }
      q_lds[t * 4 + eo] = 0.5f * q;   // scale = 1/sqrt(EMBED) folded here
      k_lds[t * 4 + eo] = k;
      vp[eo] = v;
    }
    v4[t] = vvec;
  }
  __syncthreads();

  const int lane15 = lane & 15;
  const int half   = lane >> 4;   // A/B layout: half 0 -> K=0,1 ; half 1 -> K=2,3
  const int kbase  = half * 2;

  // ---- Hoist ALL B operands (K^T tiles) out of the tile loop: j-only data ----
  v2f barr[16];
#pragma unroll
  for (int j = 0; j < 16; ++j) {
    barr[j].x = k_lds[(j * 16 + lane15) * 4 + kbase];
    barr[j].y = k_lds[(j * 16 + lane15) * 4 + kbase + 1];
  }

  float* swave = sbuf + wave * 16 * SSTRIDE;  // wave-private 16 x 256 strip

  // ---- Each wave owns row-tiles {wave, wave+4, wave+8, wave+12} ----
  for (int tile = wave; tile < SEQ / 16; tile += NWAVE) {
    const int i0 = tile * 16;

    // A operand: (scaled) Q rows i0..i0+15 (16x4, K split across lane halves)
    v2f a;
    a.x = q_lds[(i0 + lane15) * 4 + kbase];
    a.y = q_lds[(i0 + lane15) * 4 + kbase + 1];

    // ---- Pass A: 16 back-to-back WMMAs -> scores into wave-private LDS strip ----
#pragma unroll
    for (int j = 0; j < 16; ++j) {
      v8f c = {};
      c = __builtin_amdgcn_wmma_f32_16x16x4_f32(
              false, a, false, barr[j], (short)0, c, false, false);

      // C layout: VGPR r = row (r + 8*half) in this tile, col = lane15
#pragma unroll
      for (int r = 0; r < 8; ++r)
        swave[(r + 8 * half) * SSTRIDE + j * 16 + lane15] = c[r];
    }

    // ---- Pass B: row-per-lane softmax + PV (halves split the 256 columns) ----
    const int    row   = lane15;
    const float* Sr    = swave + row * SSTRIDE;
    const int    tbase = half * 128;

    float mx = -1e30f;
#pragma unroll 8
    for (int i = 0; i < 128; i += 4) {
      float4 sv = *(const float4*)(Sr + tbase + i);
      mx = fmaxf(mx, fmaxf(fmaxf(sv.x, sv.y), fmaxf(sv.z, sv.w)));
    }
    mx = fmaxf(mx, __shfl_xor(mx, 16, 32));   // combine column halves

    float sum = 0.f, o0 = 0.f, o1 = 0.f, o2 = 0.f, o3 = 0.f;
#pragma unroll 4
    for (int i = 0; i < 128; i += 4) {
      float4 sv = *(const float4*)(Sr + tbase + i);
      float p0 = __expf(sv.x - mx);
      float p1 = __expf(sv.y - mx);
      float p2 = __expf(sv.z - mx);
      float p3 = __expf(sv.w - mx);
      sum += (p0 + p1) + (p2 + p3);
      float4 va = v4[tbase + i + 0];
      float4 vb = v4[tbase + i + 1];
      float4 vc = v4[tbase + i + 2];
      float4 vd = v4[tbase + i + 3];
      o0 += p0 * va.x + p1 * vb.x + p2 * vc.x + p3 * vd.x;
      o1 += p0 * va.y + p1 * vb.y + p2 * vc.y + p3 * vd.y;
      o2 += p0 * va.z + p1 * vb.z + p2 * vc.z + p3 * vd.z;
      o3 += p0 * va.w + p1 * vb.w + p2 * vc.w + p3 * vd.w;
    }
    sum += __shfl_xor(sum, 16, 32);
    o0  += __shfl_xor(o0, 16, 32);
    o1  += __shfl_xor(o1, 16, 32);
    o2  += __shfl_xor(o2, 16, 32);
    o3  += __shfl_xor(o3, 16, 32);

    float inv = 1.f / sum;
    o0 *= inv; o1 *= inv; o2 *= inv; o3 *= inv;

    // ---- Tiny MLP: 4->8 tanh -> 4 tanh -> 1 (redundant in both halves) ----
    float h1v[H1D];
#pragma unroll
    for (int j = 0; j < H1D; ++j)
      h1v[j] = tanhf(b1[j] + o0 * W1[0 * H1D + j] + o1 * W1[1 * H1D + j]
                           + o2 * W1[2 * H1D + j] + o3 * W1[3 * H1D + j]);
    float h2v[H2D];
#pragma unroll
    for (int j = 0; j < H2D; ++j) {
      float acc = b2[j];
#pragma unroll
      for (int i = 0; i < H1D; ++i) acc += h1v[i] * W2[i * H2D + j];
      h2v[j] = tanhf(acc);
    }
    float y = b3[0];
#pragma unroll
    for (int j = 0; j < H2D; ++j) y += h2v[j] * W3[j];

    if (half == 0) out[(size_t)b * SEQ + i0 + row] = y;
  }
}

extern "C" void kernel_launch(void* const* d_in, const int* in_sizes, int n_in,
                              void* d_out, int out_size, void* d_ws, size_t ws_size,
                              hipStream_t stream) {
  const float* x  = (const float*)d_in[0];
  const float* Wq = (const float*)d_in[1];
  const float* Wk = (const float*)d_in[2];
  const float* Wv = (const float*)d_in[3];
  const float* W1 = (const float*)d_in[4];
  const float* b1 = (const float*)d_in[5];
  const float* W2 = (const float*)d_in[6];
  const float* b2 = (const float*)d_in[7];
  const float* W3 = (const float*)d_in[8];
  const float* b3 = (const float*)d_in[9];
  float* out = (float*)d_out;

  const size_t smem_bytes =
      (size_t)(SEQ * EMBED * 2 + SEQ * 4 + NWAVE * 16 * SSTRIDE) * sizeof(float);

  hybrid_attn_kernel<<<BATCH, 128, smem_bytes, stream>>>(
      x, Wq, Wk, Wv, W1, b1, W2, b2, W3, b3, out);
}